// NCLModel_91070486544456
// MI455X (gfx1250) — compile-verified
//
#include <hip/hip_runtime.h>

typedef __attribute__((ext_vector_type(2))) float v2f;
typedef __attribute__((ext_vector_type(8))) float v8f;

namespace {
constexpr int       kUsers   = 500000;
constexpr int       kItems   = 200000;
constexpr int       kNodes   = kUsers + kItems;   // 700000
constexpr int       kEmb     = 64;
constexpr int       kEdges   = 5000000;
constexpr int       kLayers  = 3;
constexpr int       kXStride = (kLayers + 1) * kEmb;              // 256 floats per node
constexpr long long kOut2Off = (long long)kUsers * kEmb;          // 32,000,000
constexpr long long kXOff    = kOut2Off + (long long)(kNodes - kItems) * kEmb; // 64,000,000
} // namespace

// ---------------- init / prep kernels ----------------

__global__ void k_zero_f32(float* __restrict__ p, int n) {
  int i = blockIdx.x * blockDim.x + threadIdx.x;
  if (i < n) p[i] = 0.0f;
}

// zero layers 1..3 of X  (192 floats per node = 48 float4)
__global__ void k_zero_xtail(float* __restrict__ X) {
  int i = blockIdx.x * blockDim.x + threadIdx.x;
  if (i >= kNodes * 48) return;
  int node = i / 48;
  int q    = i % 48;
  float4 z = make_float4(0.f, 0.f, 0.f, 0.f);
  *((float4*)(X + (long long)node * kXStride + kEmb) + q) = z;
}

// X[:,0,:] = concat(user_emb, item_emb)   (16 float4 per node)
__global__ void k_init_x0(float* __restrict__ X,
                          const float* __restrict__ user,
                          const float* __restrict__ item) {
  int i = blockIdx.x * blockDim.x + threadIdx.x;
  if (i >= kNodes * 16) return;
  int node = i >> 4;
  int q    = i & 15;
  float4 v = (node < kUsers)
                 ? *((const float4*)(user + (long long)node * kEmb) + q)
                 : *((const float4*)(item + (long long)(node - kUsers) * kEmb) + q);
  *((float4*)(X + (long long)node * kXStride) + q) = v;
}

// deg[col[e]] += 1
__global__ void k_deg(const int* __restrict__ col, float* __restrict__ deg) {
  int e = blockIdx.x * blockDim.x + threadIdx.x;
  if (e >= kEdges) return;
  __hip_atomic_fetch_add(&deg[col[e]], 1.0f, __ATOMIC_RELAXED, __HIP_MEMORY_SCOPE_AGENT);
}

// deg -> dinv in place
__global__ void k_dinv(float* __restrict__ deg) {
  int n = blockIdx.x * blockDim.x + threadIdx.x;
  if (n >= kNodes) return;
  float d = deg[n];
  deg[n] = (d > 0.0f) ? rsqrtf(d) : 0.0f;
}

// ---------------- propagation: X[:,l+1,:] = scatter_add(norm * X[row, l, :], col) ----------------
// 16 consecutive threads handle one edge (float4 chunk each): b128 coalesced gather,
// 4x global_atomic_add_f32 scatter into L2.
__global__ void k_prop(const int* __restrict__ row, const int* __restrict__ col,
                       const float* __restrict__ dinv, float* __restrict__ X, int layer) {
  int idx = blockIdx.x * blockDim.x + threadIdx.x;
  if (idx >= kEdges * 16) return;
  int e = idx >> 4;
  int c = idx & 15;
  int r  = row[e];
  int cl = col[e];
  float nm = dinv[r] * dinv[cl];
  const float4 v = *((const float4*)(X + (long long)r * kXStride + layer * kEmb) + c);
  float* dst = X + (long long)cl * kXStride + (layer + 1) * kEmb + (c << 2);
  __hip_atomic_fetch_add(dst + 0, nm * v.x, __ATOMIC_RELAXED, __HIP_MEMORY_SCOPE_AGENT);
  __hip_atomic_fetch_add(dst + 1, nm * v.y, __ATOMIC_RELAXED, __HIP_MEMORY_SCOPE_AGENT);
  __hip_atomic_fetch_add(dst + 2, nm * v.z, __ATOMIC_RELAXED, __HIP_MEMORY_SCOPE_AGENT);
  __hip_atomic_fetch_add(dst + 3, nm * v.w, __ATOMIC_RELAXED, __HIP_MEMORY_SCOPE_AGENT);
}

// ---------------- mean over layers via V_WMMA_F32_16X16X4_F32 ----------------
// One wave handles 16 nodes. For each dim d: B column n (= lane%16) holds the 4 layer
// values of node nbase+n (K rows 0/1 in lanes 0-15 across the two B VGPRs, K rows 2/3
// in lanes 16-31, matching the ISA 16x4/4x16 striping). A = all 1.0 -> D = sum over
// layers replicated across rows; lane L reads its column from c[0], scales by 0.25.
// Low half-wave writes the user slice, high half-wave writes the item slice.
__global__ void k_mean(const float* __restrict__ X,
                       float* __restrict__ out1, float* __restrict__ out2) {
  int wave = (blockIdx.x * blockDim.x + threadIdx.x) >> 5;
  int lane = threadIdx.x & 31;
  if (wave >= kNodes / 16) return;               // wave-uniform exit: EXEC stays all-1s
  int sub = lane & 15;
  int hi  = lane >> 4;
  int node = wave * 16 + sub;

  const float* rowp0 = X + (long long)node * kXStride + (hi ? 2 * kEmb : 0);
  const float* rowp1 = X + (long long)node * kXStride + (hi ? 3 * kEmb : kEmb);

  v2f a; a[0] = 1.0f; a[1] = 1.0f;
  bool w1 = (hi == 0) && (node < kUsers);
  bool w2 = (hi == 1) && (node >= kItems);

  for (int d0 = 0; d0 < kEmb; d0 += 4) {
    float4 b0 = *(const float4*)(rowp0 + d0);
    float4 b1 = *(const float4*)(rowp1 + d0);
    float bl0[4] = {b0.x, b0.y, b0.z, b0.w};
    float bl1[4] = {b1.x, b1.y, b1.z, b1.w};
    float4 res;
    float* rp = (float*)&res;
#pragma unroll
    for (int j = 0; j < 4; ++j) {
      v2f b; b[0] = bl0[j]; b[1] = bl1[j];
      v8f c = {};
      c = __builtin_amdgcn_wmma_f32_16x16x4_f32(
          /*neg_a=*/false, a, /*neg_b=*/false, b,
          /*c_mod=*/(short)0, c, /*reuse_a=*/false, /*reuse_b=*/false);
      rp[j] = c[0] * 0.25f;
    }
    if (w1) *(float4*)(out1 + (long long)node * kEmb + d0) = res;
    if (w2) *(float4*)(out2 + (long long)(node - kItems) * kEmb + d0) = res;
  }
}

// ---------------- launch ----------------

extern "C" void kernel_launch(void* const* d_in, const int* in_sizes, int n_in,
                              void* d_out, int out_size, void* d_ws, size_t ws_size,
                              hipStream_t stream) {
  (void)in_sizes; (void)n_in; (void)out_size; (void)ws_size;

  const float* user  = (const float*)d_in[0];
  const float* item  = (const float*)d_in[1];
  const int*   edges = (const int*)d_in[2];
  const int*   row   = edges;            // edge_index[0]
  const int*   col   = edges + kEdges;   // edge_index[1]

  float* out  = (float*)d_out;
  float* out1 = out;                     // out[:NUM_USERS]
  float* out2 = out + kOut2Off;          // out[NUM_ITEMS:]
  float* X    = out + kXOff;             // all_embeddings [N, 4, 64] — used as working state
  float* deg  = (float*)d_ws;            // kNodes floats (deg, then dinv in place)

  constexpr int B = 256;
  auto cdiv = [](long long a, long long b) { return (int)((a + b - 1) / b); };

  k_zero_f32 <<<cdiv(kNodes, B),            B, 0, stream>>>(deg, kNodes);
  k_zero_xtail<<<cdiv((long long)kNodes * 48, B), B, 0, stream>>>(X);
  k_init_x0  <<<cdiv((long long)kNodes * 16, B),  B, 0, stream>>>(X, user, item);
  k_deg      <<<cdiv(kEdges, B),            B, 0, stream>>>(col, deg);
  k_dinv     <<<cdiv(kNodes, B),            B, 0, stream>>>(deg);

  for (int l = 0; l < kLayers; ++l) {
    k_prop <<<cdiv((long long)kEdges * 16, B), B, 0, stream>>>(row, col, deg, X, l);
  }

  int waves = kNodes / 16; // 43750, exact
  k_mean <<<cdiv((long long)waves * 32, B), B, 0, stream>>>(X, out1, out2);
}